// DictionaryMatchingTv_266287972748
// MI455X (gfx1250) — compile-verified
//
#include <hip/hip_runtime.h>
#include <hip/hip_bf16.h>

// MI455X (gfx1250) dictionary-matching + TV objective.
// Latency-bound problem (~5MB HBM traffic, ~7 GFLOP f32): keep full f32
// precision and use V_WMMA_F32_16X16X4_F32 for the dual contractions
// (dot = D*u, n2 = D^2 * v) with per-bin delta masking on the B operand.

typedef float v2f __attribute__((ext_vector_type(2)));
typedef float v8f __attribute__((ext_vector_type(8)));

constexpr int NX   = 256;
constexpr int NY   = 256;
constexpr int NPIX = NX * NY;       // 65536
constexpr int ETL  = 16;
constexpr int NT2  = 100;
constexpr int NB1  = 30;
constexpr int TPAD = 112;           // 7 t-tiles of 16
constexpr int NTILE = NPIX / 16;    // 4096 pixel tiles

__global__ __launch_bounds__(224) void k_tile(
    const float* __restrict__ sig,   // (NPIX, ETL)
    const float* __restrict__ db,    // (NT2, NB1, ETL)
    const float* __restrict__ b1s,   // (NB1)
    const float* __restrict__ dtt,   // (ETL)
    const float* __restrict__ est,   // (2, NPIX)
    float* __restrict__ tv,          // (NPIX) out
    float* __restrict__ partials)    // (NTILE, TPAD) out
{
    __shared__ __align__(16) float sU[16][ETL];   // eta * sig_hat
    __shared__ __align__(16) float sV[16][ETL];   // eta^2
    __shared__ float sSz[16];                      // ||sig|| > 0 ? 1 : 0
    __shared__ int   sIdx[16];                     // b1 bin index

    const int tid = threadIdx.x;
    const int p0  = blockIdx.x * 16;

    // ---- Stage 1: per-pixel prep (16 pixels, threads 0..15) ----
    if (tid < 16) {
        const int p = p0 + tid;
        const float* sp = sig + (size_t)p * ETL;
        float s[ETL];
        float nrm = 0.f;
#pragma unroll
        for (int e = 0; e < ETL; ++e) { s[e] = sp[e]; nrm += s[e] * s[e]; }
        const float inv = (nrm > 0.f) ? rsqrtf(nrm) : 0.f;
        sSz[tid] = (nrm > 0.f) ? 1.f : 0.f;

        const float t2p = 1.f + 499.f * est[p];          // T2P_RANGE
#pragma unroll
        for (int e = 0; e < ETL; ++e) {
            const float eta = expf(-dtt[e] / t2p);
            sU[tid][e] = s[e] * inv * eta;
            sV[tid][e] = eta * eta;
        }

        const float b1 = 0.2f + 1.4f * est[NPIX + p];    // B1_RANGE
        int best = 0; float bd = 3.4e38f;
        for (int k = 0; k < NB1; ++k) {
            float d = b1 - b1s[k]; d *= d;
            if (d < bd) { bd = d; best = k; }             // first-min, like argmin
        }
        sIdx[tid] = best;

        // TV term (jnp.gradient, spacing 1)
        const int i = p >> 8, j = p & (NY - 1);
        auto bval = [&](int q) { return 0.2f + 1.4f * est[NPIX + q]; };
        float g0 = (i == 0)      ? (bval(p + NY) - b1)
                 : (i == NX - 1) ? (b1 - bval(p - NY))
                                 : 0.5f * (bval(p + NY) - bval(p - NY));
        float g1 = (j == 0)      ? (bval(p + 1) - b1)
                 : (j == NY - 1) ? (b1 - bval(p - 1))
                                 : 0.5f * (bval(p + 1) - bval(p - 1));
        tv[p] = fabsf(g0) + fabsf(g1);
    }
    __syncthreads();

    // ---- Stage 2: WMMA GEMM, one t-tile (16 rows) per wave ----
    const int wv   = tid >> 5;        // 0..6
    const int lane = tid & 31;
    const int half = lane >> 4;       // 0: lanes 0-15, 1: lanes 16-31
    const int lr   = lane & 15;
    const int t0   = wv * 16;
    const int trow = t0 + lr;
    const float validf = (trow < NT2) ? 1.f : 0.f;
    const int trow_c   = (trow < NT2) ? trow : (NT2 - 1);   // clamp, avoid OOB

    // bins present in this 16-pixel tile
    unsigned binmask = 0;
#pragma unroll
    for (int i = 0; i < 16; ++i) binmask |= 1u << sIdx[i];
    const int mybin = sIdx[lr];

    // B-operand base fragments (shared across bins): lane = column n = lr,
    // VGPR j holds K = ks*4 + 2*half + j  (ISA 7.12.2 layout)
    v2f uf[4], vf[4];
#pragma unroll
    for (int ks = 0; ks < 4; ++ks) {
        const int e0 = ks * 4 + 2 * half;
        uf[ks] = *(const v2f*)&sU[lr][e0];
        vf[ks] = *(const v2f*)&sV[lr][e0];
    }

    v8f accD = {0.f, 0.f, 0.f, 0.f, 0.f, 0.f, 0.f, 0.f};
    v8f accN = {0.f, 0.f, 0.f, 0.f, 0.f, 0.f, 0.f, 0.f};

    const float* dbrow = db + (size_t)trow_c * NB1 * ETL;

    for (int b = 0; b < NB1; ++b) {
        if (!((binmask >> b) & 1u)) continue;             // skip all-zero K-blocks
        const float m = (mybin == b) ? 1.f : 0.f;         // delta mask on columns
        const float* ab = dbrow + b * ETL;
#pragma unroll
        for (int ks = 0; ks < 4; ++ks) {
            const int e0 = ks * 4 + 2 * half;
            v2f a = *(const v2f*)(ab + e0);
            a *= validf;                                   // zero padded t rows
            v2f bu = uf[ks] * m;
            v2f bw = vf[ks] * m;
            accD = __builtin_amdgcn_wmma_f32_16x16x4_f32(
                false, a, false, bu, (short)0, accD, false, false);
            v2f a2 = a * a;                                // D^2 fragment in VGPRs
            accN = __builtin_amdgcn_wmma_f32_16x16x4_f32(
                false, a2, false, bw, (short)0, accN, false, false);
        }
    }

    // ---- Stage 3: distance + reduce over the 16 pixel columns ----
    const float sz = sSz[lr];
#pragma unroll
    for (int r = 0; r < 8; ++r) {
        const int t = t0 + r + 8 * half;                  // C layout: M = r + 8*half
        float n2 = accN[r];
        float dp = accD[r];
        float d2 = 0.f;
        if (t < NT2) {
            const float invn = (n2 > 0.f) ? rsqrtf(n2) : 0.f;
            d2 = ((n2 > 0.f) ? 1.f : 0.f) + sz - 2.f * dp * invn;
        }
        d2 += __shfl_xor(d2, 1, 16);
        d2 += __shfl_xor(d2, 2, 16);
        d2 += __shfl_xor(d2, 4, 16);
        d2 += __shfl_xor(d2, 8, 16);
        if (lr == 0) partials[(size_t)blockIdx.x * TPAD + t] = d2;
    }
}

__global__ __launch_bounds__(256) void k_reduce_t(
    const float* __restrict__ partials, float* __restrict__ acc)
{
    __shared__ float red[256];
    const int t = blockIdx.x;                  // 0..TPAD-1
    float s = 0.f;
    for (int i = threadIdx.x; i < NTILE; i += 256)
        s += partials[(size_t)i * TPAD + t];
    red[threadIdx.x] = s;
    __syncthreads();
    for (int st = 128; st > 0; st >>= 1) {
        if (threadIdx.x < st) red[threadIdx.x] += red[threadIdx.x + st];
        __syncthreads();
    }
    if (threadIdx.x == 0) acc[t] = red[0];
}

__global__ __launch_bounds__(256) void k_final(
    const float* __restrict__ tv, const float* __restrict__ acc,
    float* __restrict__ out)
{
    __shared__ float red[256];
    float s = 0.f;
    for (int i = threadIdx.x; i < NPIX; i += 256) s += tv[i];      // f_2
    if (threadIdx.x < NT2) s += sqrtf(fmaxf(acc[threadIdx.x], 0.f)); // f_1
    red[threadIdx.x] = s;
    __syncthreads();
    for (int st = 128; st > 0; st >>= 1) {
        if (threadIdx.x < st) red[threadIdx.x] += red[threadIdx.x + st];
        __syncthreads();
    }
    if (threadIdx.x == 0) out[0] = red[0];
}

extern "C" void kernel_launch(void* const* d_in, const int* in_sizes, int n_in,
                              void* d_out, int out_size, void* d_ws, size_t ws_size,
                              hipStream_t stream) {
    const float* sig = (const float*)d_in[0];   // (256,256,16)
    const float* db  = (const float*)d_in[1];   // (100,30,16)
    // d_in[2] = db_t2s_ms (unused by the objective)
    const float* b1s = (const float*)d_in[3];   // (30)
    const float* dtt = (const float*)d_in[4];   // (16)
    const float* est = (const float*)d_in[5];   // (2,256,256)
    float* out = (float*)d_out;

    float* ws       = (float*)d_ws;
    float* tv       = ws;                        // NPIX floats
    float* partials = ws + NPIX;                 // NTILE*TPAD floats
    float* acc      = partials + (size_t)NTILE * TPAD; // TPAD floats

    k_tile<<<NTILE, 224, 0, stream>>>(sig, db, b1s, dtt, est, tv, partials);
    k_reduce_t<<<TPAD, 256, 0, stream>>>(partials, acc);
    k_final<<<1, 256, 0, stream>>>(tv, acc, out);
}